// BinarizeLayer_41154376631026
// MI455X (gfx1250) — compile-verified
//
#include <hip/hip_runtime.h>
#include <stdint.h>

#define NBINS 256
#define WS_HIST_OFF 16
#define WS_GRAY_OFF 4096

#if defined(__has_builtin)
#if __has_builtin(__builtin_amdgcn_wmma_f32_16x16x4_f32)
#define HAVE_WMMA_F32X4 1
#endif
#endif

typedef __attribute__((ext_vector_type(2))) float v2f;
typedef __attribute__((ext_vector_type(8))) float v8f;

__device__ __forceinline__ unsigned flipf(float f) {
    unsigned u = __float_as_uint(f);
    return (u & 0x80000000u) ? ~u : (u | 0x80000000u);
}
__device__ __forceinline__ float unflipf(unsigned u) {
    return __uint_as_float((u & 0x80000000u) ? (u & 0x7fffffffu) : ~u);
}
__device__ __forceinline__ float gray3(float r, float g, float b) {
    return fmaf(0.2989f, r, fmaf(0.5870f, g, 0.1140f * b));
}

// ---------------- kernel 0: init scratch ----------------
__global__ void k_init(unsigned* mm, unsigned* hist, float* thresh) {
    int t = threadIdx.x;
    if (t == 0) { mm[0] = 0xFFFFFFFFu; mm[1] = 0u; *thresh = 0.0f; }
    if (t < NBINS) hist[t] = 0u;
}

// ------- kernel 1: RGB->gray via async-to-LDS staging, global min/max -------
__global__ __launch_bounds__(256) void k_gray_minmax(
    const float* __restrict__ in, float* __restrict__ gray,
    unsigned* mm, unsigned npix) {
    __shared__ float4 tile4[256 * 3];           // 12 KB staging: 4 pixels/thread
    __shared__ unsigned redmin[8], redmax[8];
    const unsigned t = threadIdx.x;
    const unsigned stride = gridDim.x * blockDim.x;
    const unsigned ngrp = npix >> 2;            // groups of 4 pixels
    const bool useG = (gray != nullptr);

    float vmin = 3.4e38f, vmax = -3.4e38f;
    unsigned ldsoff = (unsigned)(uintptr_t)(&tile4[t * 3]);

    for (unsigned base = blockIdx.x * blockDim.x + t; base < ngrp; base += stride) {
        const float* gp = in + (size_t)base * 12;   // 4 pixels * 3 channels
        // CDNA5 async global->LDS path (3 x b128 per lane), then wait ASYNCcnt.
        asm volatile(
            "global_load_async_to_lds_b128 %0, %1, off\n\t"
            "global_load_async_to_lds_b128 %0, %1, off offset:16\n\t"
            "global_load_async_to_lds_b128 %0, %1, off offset:32\n\t"
            "s_wait_asynccnt 0"
            :: "v"(ldsoff), "v"(gp) : "memory");
        float4 a = tile4[t * 3 + 0];
        float4 b = tile4[t * 3 + 1];
        float4 c = tile4[t * 3 + 2];
        float y0 = gray3(a.x, a.y, a.z);
        float y1 = gray3(a.w, b.x, b.y);
        float y2 = gray3(b.z, b.w, c.x);
        float y3 = gray3(c.y, c.z, c.w);
        vmin = fminf(vmin, fminf(fminf(y0, y1), fminf(y2, y3)));
        vmax = fmaxf(vmax, fmaxf(fmaxf(y0, y1), fmaxf(y2, y3)));
        if (useG) {
            float4 o; o.x = y0; o.y = y1; o.z = y2; o.w = y3;
            ((float4*)gray)[base] = o;
        }
    }
    // wave32 reduction then cross-wave via LDS, one atomic pair per block
    for (int m = 16; m >= 1; m >>= 1) {
        vmin = fminf(vmin, __shfl_xor(vmin, m, 32));
        vmax = fmaxf(vmax, __shfl_xor(vmax, m, 32));
    }
    int wid = t >> 5, lid = t & 31;
    if (lid == 0) { redmin[wid] = flipf(vmin); redmax[wid] = flipf(vmax); }
    __syncthreads();
    if (t == 0) {
        unsigned mn = redmin[0], mx = redmax[0];
        for (int i = 1; i < 8; i++) { mn = min(mn, redmin[i]); mx = max(mx, redmax[i]); }
        atomicMin(&mm[0], mn);
        atomicMax(&mm[1], mx);
    }
}

// ---------------- kernel 2: histogram (LDS-privatized) ----------------
__global__ __launch_bounds__(256) void k_hist(
    const float* __restrict__ gray, const float* __restrict__ in,
    const unsigned* mm, unsigned* hist, unsigned npix) {
    __shared__ unsigned lh[NBINS];
    const unsigned t = threadIdx.x;
    lh[t] = 0u;
    __syncthreads();
    const float mn = unflipf(mm[0]);
    const float mx = unflipf(mm[1]);
    const float width = (mx - mn) / 256.0f;
    const unsigned stride = gridDim.x * blockDim.x;
    const unsigned ngrp = npix >> 2;
    const bool useG = (gray != nullptr);

    for (unsigned base = blockIdx.x * blockDim.x + t; base < ngrp; base += stride) {
        float y0, y1, y2, y3;
        if (useG) {
            float4 g = ((const float4*)gray)[base];
            y0 = g.x; y1 = g.y; y2 = g.z; y3 = g.w;
        } else {
            const float4* ip = (const float4*)(in + (size_t)base * 12);
            float4 a = ip[0], b = ip[1], c = ip[2];
            y0 = gray3(a.x, a.y, a.z);
            y1 = gray3(a.w, b.x, b.y);
            y2 = gray3(b.z, b.w, c.x);
            y3 = gray3(c.y, c.z, c.w);
        }
        int i0 = min(max((int)((y0 - mn) / width), 0), 255);
        int i1 = min(max((int)((y1 - mn) / width), 0), 255);
        int i2 = min(max((int)((y2 - mn) / width), 0), 255);
        int i3 = min(max((int)((y3 - mn) / width), 0), 255);
        atomicAdd(&lh[i0], 1u);
        atomicAdd(&lh[i1], 1u);
        atomicAdd(&lh[i2], 1u);
        atomicAdd(&lh[i3], 1u);
    }
    __syncthreads();
    unsigned v = lh[t];
    if (v) atomicAdd(&hist[t], v);
}

// -------- kernel 3: Otsu scan; 256-long prefix sums via WMMA (16x16x4 f32) ---
// P = M * U (U upper-triangular ones) gives row-wise inclusive scans of the
// 16x16 reshaped histogram; a 16-row offset fixup yields the full cumsum.
__global__ void k_otsu(const unsigned* hist, const unsigned* mm, float* thresh) {
    __shared__ float Wc[256], Mc[256], SP[256], SQ[256];
    __shared__ float bval[32];
    __shared__ int bidx[32];
    const int lane = threadIdx.x;   // single wave32
    const float mn = unflipf(mm[0]);
    const float mx = unflipf(mm[1]);
    const float width = (mx - mn) / 256.0f;

    for (int i = 0; i < 8; i++) {
        int idx = lane * 8 + i;
        float c = (float)hist[idx];
        Wc[idx] = c;
        Mc[idx] = c * (mn + width * ((float)idx + 0.5f));
    }
    __syncthreads();

#ifdef HAVE_WMMA_F32X4
    {
        const int l = lane & 15, h = lane >> 4;
        v8f P = {}; v8f Q = {};
#pragma unroll
        for (int kb = 0; kb < 4; kb++) {
            int k0 = 4 * kb + 2 * h;      // per-ISA 32-bit A/B lane->K mapping
            v2f a, am, b;
            a.x  = Wc[16 * l + k0]; a.y  = Wc[16 * l + k0 + 1];
            am.x = Mc[16 * l + k0]; am.y = Mc[16 * l + k0 + 1];
            b.x = (k0     <= l) ? 1.0f : 0.0f;   // U[k][n], n = l
            b.y = (k0 + 1 <= l) ? 1.0f : 0.0f;
            P = __builtin_amdgcn_wmma_f32_16x16x4_f32(false, a, false, b, (short)0, P, false, false);
            Q = __builtin_amdgcn_wmma_f32_16x16x4_f32(false, am, false, b, (short)0, Q, false, false);
        }
        // C/D layout: VGPR v -> row (v + 8*half), col = l
#pragma unroll
        for (int v = 0; v < 8; v++) {
            SP[16 * (v + 8 * h) + l] = P[v];
            SQ[16 * (v + 8 * h) + l] = Q[v];
        }
    }
    __syncthreads();
    if (lane == 0) {  // carry row totals forward (15 * 16 adds, negligible)
        for (int r = 1; r < 16; r++) {
            float ow = SP[16 * (r - 1) + 15];
            float oq = SQ[16 * (r - 1) + 15];
            for (int c = 0; c < 16; c++) {
                SP[16 * r + c] += ow;
                SQ[16 * r + c] += oq;
            }
        }
    }
#else
    if (lane == 0) {
        float s = 0.0f, q = 0.0f;
        for (int i = 0; i < 256; i++) {
            s += Wc[i]; SP[i] = s;
            q += Mc[i]; SQ[i] = q;
        }
    }
#endif
    __syncthreads();

    const float Wt = SP[255], Qt = SQ[255];
    float best = -1.0f; int bi = 0;
    for (int i = 0; i < 8; i++) {
        int tt = lane + 32 * i;
        if (tt < 255) {
            float w1 = SP[tt], s1 = SQ[tt];
            float w2 = Wt - w1, s2 = Qt - s1;      // reverse-cumsum identities
            float m1 = s1 / fmaxf(w1, 1.0f);
            float m2 = s2 / fmaxf(w2, 1.0f);
            float d = m1 - m2;
            float var = w1 * w2 * d * d;
            if (var > best) { best = var; bi = tt; }
        }
    }
    bval[lane] = best; bidx[lane] = bi;
    __syncthreads();
    if (lane == 0) {
        float bb = bval[0]; int ii = bidx[0];
        for (int k = 1; k < 32; k++) {
            if (bval[k] > bb || (bval[k] == bb && bidx[k] < ii)) { bb = bval[k]; ii = bidx[k]; }
        }
        *thresh = mn + width * ((float)ii + 0.5f);   // first-argmax center
    }
}

// ---------------- kernel 4: binarize ----------------
__global__ __launch_bounds__(256) void k_bin(
    const float* __restrict__ gray, const float* __restrict__ in,
    const float* thresh, float* __restrict__ out, unsigned npix) {
    const float th = *thresh;
    const unsigned stride = gridDim.x * blockDim.x;
    const unsigned ngrp = npix >> 2;
    const bool useG = (gray != nullptr);
    for (unsigned base = blockIdx.x * blockDim.x + threadIdx.x; base < ngrp; base += stride) {
        float y0, y1, y2, y3;
        if (useG) {
            float4 g = ((const float4*)gray)[base];
            y0 = g.x; y1 = g.y; y2 = g.z; y3 = g.w;
        } else {
            const float4* ip = (const float4*)(in + (size_t)base * 12);
            float4 a = ip[0], b = ip[1], c = ip[2];
            y0 = gray3(a.x, a.y, a.z);
            y1 = gray3(a.w, b.x, b.y);
            y2 = gray3(b.z, b.w, c.x);
            y3 = gray3(c.y, c.z, c.w);
        }
        float4 o;
        o.x = (y0 > th) ? 1.0f : 0.0f;
        o.y = (y1 > th) ? 1.0f : 0.0f;
        o.z = (y2 > th) ? 1.0f : 0.0f;
        o.w = (y3 > th) ? 1.0f : 0.0f;
        ((float4*)out)[base] = o;
    }
}

extern "C" void kernel_launch(void* const* d_in, const int* in_sizes, int n_in,
                              void* d_out, int out_size, void* d_ws, size_t ws_size,
                              hipStream_t stream) {
    const float* in = (const float*)d_in[0];
    float* out = (float*)d_out;
    unsigned char* ws = (unsigned char*)d_ws;
    unsigned* mm = (unsigned*)ws;                  // [0]=min(flipped), [1]=max(flipped)
    float* thresh = (float*)(ws + 8);
    unsigned* hist = (unsigned*)(ws + WS_HIST_OFF);
    const unsigned npix = (unsigned)(in_sizes[0] / 3);

    // Keep gray intermediate in workspace (it then lives in the 192MB L2 for
    // passes 2 and 4, saving ~256 MiB of HBM re-reads). Deterministic branch.
    size_t need = (size_t)WS_GRAY_OFF + (size_t)npix * sizeof(float);
    float* gray = (ws_size >= need) ? (float*)(ws + WS_GRAY_OFF) : nullptr;

    k_init<<<1, 256, 0, stream>>>(mm, hist, thresh);
    k_gray_minmax<<<2048, 256, 0, stream>>>(in, gray, mm, npix);
    k_hist<<<2048, 256, 0, stream>>>(gray, in, mm, hist, npix);
    k_otsu<<<1, 32, 0, stream>>>(hist, mm, thresh);
    k_bin<<<2048, 256, 0, stream>>>(gray, in, thresh, out, npix);
}